// DissipativeHamiltonianDerivation_77979426226467
// MI455X (gfx1250) — compile-verified
//
#include <hip/hip_runtime.h>
#include <math.h>

#define NA 1536
#define NT 96   // 16-wide tiles
#define LSTR 20 // LDS row stride (floats), 16B aligned, conflict-free

typedef float v2f __attribute__((ext_vector_type(2)));
typedef float v8f __attribute__((ext_vector_type(8)));

__device__ __forceinline__ float sig_(float x) { return 1.f / (1.f + __expf(-x)); }
__device__ __forceinline__ float sp_(float x) { return (x > 20.f) ? x : log1pf(__expf(x)); }
// g-factor: f'(softplus(s)) * sigmoid(s), f'(d) = (d-2)/d^3 ; s >= 0 always here
__device__ __forceinline__ float gfun_(float s) {
  float d = sp_(s);
  return (d - 2.f) / (d * d * d) * sig_(s);
}

// ---------------- Kernel 1: per-node precompute + dq output ----------------
__global__ __launch_bounds__(256) void hd_precompute(
    const float* __restrict__ v, const float* __restrict__ m, const float* __restrict__ p,
    const float* __restrict__ q, const float* __restrict__ mvw,
    const float* __restrict__ W_T, const float* __restrict__ W1_w,
    const float* __restrict__ W1_b, const float* __restrict__ W_F,
    float* __restrict__ R, float* __restrict__ alpha, float* __restrict__ beta,
    float* __restrict__ mol, float* __restrict__ mD, float* __restrict__ dq_out)
{
  int i = blockIdx.x * blockDim.x + threadIdx.x;
  if (i >= NA) return;
  float mi = m[i];

  float Racc[16], zT[16], zF[16];
#pragma unroll
  for (int h = 0; h < 16; ++h) { Racc[h] = 0.f; zT[h] = 0.f; zF[h] = 0.f; }

  for (int c = 0; c < 64; ++c) {
    float vc = sig_(v[i * 64 + c]);
#pragma unroll
    for (int h = 0; h < 16; ++h) {
      Racc[h] += vc * W1_w[h * 96 + c];
      zT[h]   += vc * W_T[h * 96 + c];
    }
  }
  for (int k = 0; k < 32; ++k) {
    float qk = q[i * 32 + k];
    float pk = p[i * 32 + k];
#pragma unroll
    for (int h = 0; h < 16; ++h) {
      Racc[h] += qk * W1_w[h * 96 + 64 + k];
      zT[h]   += pk * W_T[h * 96 + 64 + k];
      zF[h]   += pk * W_F[h * 32 + k];
    }
  }

  float nrm = 0.f, rb = 0.f, bb = 0.f;
#pragma unroll
  for (int h = 0; h < 16; ++h) {
    float bh = W1_b[h];
    R[i * 16 + h] = Racc[h];
    nrm += Racc[h] * Racc[h];
    rb  += Racc[h] * bh;
    bb  += bh * bh;
  }
  alpha[i] = nrm - 2.f * rb;        // s_ij = alpha_i + beta_j - 2 R_i.R_j
  beta[i]  = nrm + 2.f * rb + bb;

  float coefT[16], coefF[16];
#pragma unroll
  for (int h = 0; h < 16; ++h) {
    coefT[h] = 2.f * sp_(zT[h]) * sig_(zT[h]) / mi;
    coefF[h] = 2.f * sp_(zF[h]) * sig_(zF[h]) / (mi * mi);
  }
  for (int k = 0; k < 32; ++k) {
    float aT = 0.f, aF = 0.f;
#pragma unroll
    for (int h = 0; h < 16; ++h) {
      aT += coefT[h] * W_T[h * 96 + 64 + k];
      aF += coefF[h] * W_F[h * 32 + k];
    }
    dq_out[i * 32 + k] = aT;        // dq = dH/dp
    mD[i * 32 + k]     = mi * aF;   // m * dD/dp (used in dp)
  }

  // recover molecule id from one-hot mvw [48, NA]
  float mo = 0.f;
  for (int r = 0; r < 48; ++r) mo += (float)r * mvw[r * NA + i];
  mol[i] = mo;
}

// ---------------- Kernel 2: pairwise potential gradient via f32 WMMA ----------------
__global__ __launch_bounds__(128) void hd_pairwise(
    const float* __restrict__ R, const float* __restrict__ alpha, const float* __restrict__ beta,
    const float* __restrict__ mol, const float* __restrict__ m, const float* __restrict__ mD,
    const float* __restrict__ W1_w, const float* __restrict__ W1_b,
    float* __restrict__ dp_out)
{
  __shared__ float lRI[16 * LSTR];
  __shared__ float lRJ[4][16 * LSTR];
  __shared__ float lG [4][16 * LSTR];
  __shared__ float lGp[4][16 * LSTR];
  __shared__ float redGR [4][16][16];
  __shared__ float redGTR[4][16][16];
  __shared__ float redGr[4][16];
  __shared__ float redGc[4][16];

  const int iBase = blockIdx.x * 16;
  const int tid   = threadIdx.x;
  const int lane  = tid & 31;
  const int wv    = tid >> 5;
  const int half  = lane >> 4;
  const int mrowA = lane & 15;   // A-matrix row index (ISA 16x4 f32 layout)
  const int ncol  = lane & 15;   // C/D & B column (N) index

  for (int idx = tid; idx < 256; idx += 128)
    lRI[(idx >> 4) * LSTR + (idx & 15)] = R[iBase * 16 + idx];
  __syncthreads();

  // Hoist A fragments of R_I (A[m][k=h]); lanes0-15: K=k0,k0+1; lanes16-31: K=k0+2,k0+3
  v2f aRI[4];
#pragma unroll
  for (int c = 0; c < 4; ++c) {
    int k = c * 4 + 2 * half;
    v2f a; a.x = lRI[mrowA * LSTR + k]; a.y = lRI[mrowA * LSTR + k + 1];
    aRI[c] = a;
  }

  // Per-lane loop-invariant row data (C-layout rows: M = r + 8*half)
  float aP[8], bP[8], moP[8], mP[8];
#pragma unroll
  for (int r = 0; r < 8; ++r) {
    int gi = iBase + r + 8 * half;
    aP[r] = alpha[gi]; bP[r] = beta[gi]; moP[r] = mol[gi]; mP[r] = m[gi];
  }

  float accGr[8], accGc[8];
#pragma unroll
  for (int r = 0; r < 8; ++r) { accGr[r] = 0.f; accGc[r] = 0.f; }
  v8f cGR = {}; v8f cGTR = {};

  float* RJ = lRJ[wv];
  float* G  = lG[wv];
  float* Gp = lGp[wv];

  for (int jt = wv; jt < NT; jt += 4) {
    const int jBase = jt * 16;
#pragma unroll
    for (int t = 0; t < 8; ++t) {
      int idx = lane * 8 + t;
      RJ[(idx >> 4) * LSTR + (idx & 15)] = R[jBase * 16 + idx];
    }
    const int j = jBase + ncol;
    float a_j = alpha[j], b_j = beta[j], mo_j = mol[j], m_j = m[j];
    __asm volatile("s_wait_dscnt 0" ::: "memory");

    // S[i][j] = sum_h R_I[i,h] * R_J[j,h]   (B[k=h][n=j] = R_J[j][h])
    v8f S = {};
#pragma unroll
    for (int c = 0; c < 4; ++c) {
      int k = c * 4 + 2 * half;
      v2f b; b.x = RJ[ncol * LSTR + k]; b.y = RJ[ncol * LSTR + k + 1];
      S = __builtin_amdgcn_wmma_f32_16x16x4_f32(false, aRI[c], false, b, (short)0, S, false, false);
    }

    // Elementwise: g_ij (rows) and g_ji (for transposed sums), masked
#pragma unroll
    for (int r = 0; r < 8; ++r) {
      int mrow = r + 8 * half;
      int gi = iBase + mrow;
      float dot = S[r];
      float s1 = aP[r] + b_j - 2.f * dot;  // s_{i,j}
      float s2 = a_j + bP[r] - 2.f * dot;  // s_{j,i}
      bool same = (moP[r] == mo_j) && (gi != j);
      float mm = same ? (mP[r] * m_j) : 0.f;
      float g1 = mm * gfun_(s1);
      float g2 = mm * gfun_(s2);
      accGr[r] += g1;                      // Gr_i  = sum_j g_ij
      accGc[r] += g2;                      // Gc_i  = sum_j g_ji
      G [mrow * LSTR + ncol] = g1;
      Gp[mrow * LSTR + ncol] = g2;
    }
    __asm volatile("s_wait_dscnt 0" ::: "memory");

    // GR  += G  @ R_J   ;   GTR += G' @ R_J   (B[k=jl][n=h] = R_J[jl][h])
#pragma unroll
    for (int c = 0; c < 4; ++c) {
      int k = c * 4 + 2 * half;
      v2f b;  b.x  = RJ[k * LSTR + ncol];       b.y  = RJ[(k + 1) * LSTR + ncol];
      v2f ag; ag.x = G [mrowA * LSTR + k];      ag.y = G [mrowA * LSTR + k + 1];
      v2f ap; ap.x = Gp[mrowA * LSTR + k];      ap.y = Gp[mrowA * LSTR + k + 1];
      cGR  = __builtin_amdgcn_wmma_f32_16x16x4_f32(false, ag, false, b, (short)0, cGR,  false, false);
      cGTR = __builtin_amdgcn_wmma_f32_16x16x4_f32(false, ap, false, b, (short)0, cGTR, false, false);
    }
  }

  // Cross-lane reduce Gr/Gc within each 16-lane half (rows live across lanes)
#pragma unroll
  for (int r = 0; r < 8; ++r) {
    float s1 = accGr[r], s2 = accGc[r];
#pragma unroll
    for (int off = 1; off < 16; off <<= 1) {
      s1 += __shfl_xor(s1, off, 32);
      s2 += __shfl_xor(s2, off, 32);
    }
    if (ncol == 0) {
      redGr[wv][r + 8 * half] = s1;
      redGc[wv][r + 8 * half] = s2;
    }
  }
#pragma unroll
  for (int r = 0; r < 8; ++r) {
    int mrow = r + 8 * half;
    redGR [wv][mrow][ncol] = cGR[r];
    redGTR[wv][mrow][ncol] = cGTR[r];
  }
  __syncthreads();

  // Epilogue: A_l[h] = R_l[h](Gc+Gr) + b_h(Gc-Gr) - GTR - GR ; dp = -(2 A@Wq + m*dDdp)
  int lr = tid >> 3;
  int k0 = (tid & 7) * 4;
  int gi = iBase + lr;
  float Gr = redGr[0][lr] + redGr[1][lr] + redGr[2][lr] + redGr[3][lr];
  float Gc = redGc[0][lr] + redGc[1][lr] + redGc[2][lr] + redGc[3][lr];
  float A[16];
#pragma unroll
  for (int h = 0; h < 16; ++h) {
    float grh = redGR [0][lr][h] + redGR [1][lr][h] + redGR [2][lr][h] + redGR [3][lr][h];
    float gth = redGTR[0][lr][h] + redGTR[1][lr][h] + redGTR[2][lr][h] + redGTR[3][lr][h];
    A[h] = lRI[lr * LSTR + h] * (Gc + Gr) + W1_b[h] * (Gc - Gr) - gth - grh;
  }
#pragma unroll
  for (int kk = 0; kk < 4; ++kk) {
    int k = k0 + kk;
    float acc = 0.f;
#pragma unroll
    for (int h = 0; h < 16; ++h) acc += A[h] * W1_w[h * 96 + 64 + k];
    dp_out[gi * 32 + k] = -(2.f * acc + mD[gi * 32 + k]);
  }
}

extern "C" void kernel_launch(void* const* d_in, const int* in_sizes, int n_in,
                              void* d_out, int out_size, void* d_ws, size_t ws_size,
                              hipStream_t stream) {
  (void)in_sizes; (void)n_in; (void)out_size; (void)ws_size;
  const float* v    = (const float*)d_in[0];
  // d_in[1] = e: unused downstream in the reference
  const float* m    = (const float*)d_in[2];
  const float* p    = (const float*)d_in[3];
  const float* q    = (const float*)d_in[4];
  const float* mvw  = (const float*)d_in[5];
  const float* W_T  = (const float*)d_in[6];
  const float* W1_w = (const float*)d_in[7];
  const float* W1_b = (const float*)d_in[8];
  const float* W_F  = (const float*)d_in[9];

  float* out = (float*)d_out;
  float* dp_out = out;            // (dp, dq) concatenated
  float* dq_out = out + NA * 32;

  float* ws    = (float*)d_ws;
  float* R     = ws;              // NA*16
  float* alpha = R + NA * 16;     // NA
  float* beta  = alpha + NA;      // NA
  float* mol   = beta + NA;       // NA
  float* mD    = mol + NA;        // NA*32

  hd_precompute<<<NA / 256, 256, 0, stream>>>(v, m, p, q, mvw, W_T, W1_w, W1_b, W_F,
                                              R, alpha, beta, mol, mD, dq_out);
  hd_pairwise<<<NT, 128, 0, stream>>>(R, alpha, beta, mol, m, mD, W1_w, W1_b, dp_out);
}